// ECE_50809463112240
// MI455X (gfx1250) — compile-verified
//
#include <hip/hip_runtime.h>
#include <hip/hip_bf16.h>
#include <stdint.h>

// ECE on MI455X (gfx1250): memory-bound streaming reduction.
// Data path: TDM (tensor_load_to_lds) DMA of preds tiles into LDS, per-wave
// double buffering synchronized with s_wait_tensorcnt; ds_add_f32 LDS bin
// accumulators; global_atomic_add_f32 block flush; tiny finalize kernel.

#define N_BINS 15
#define ROW_C 10
#define CHUNK_ROWS 64
#define CHUNK_ELTS (CHUNK_ROWS * ROW_C)   // 640 f32 = 2560 bytes per TDM tile
#define WAVES_PER_BLOCK 8
#define CPW 4                             // chunks per wave (compile-time)
#define ROWS_PER_BLOCK (CHUNK_ROWS * CPW * WAVES_PER_BLOCK)  // 2048

typedef __attribute__((ext_vector_type(4))) unsigned int u32x4;
typedef __attribute__((ext_vector_type(8))) unsigned int u32x8;

// ---------------------------------------------------------------------------
// Per-row stats: max, first-occurrence argmax, bin index replicating
// jnp.searchsorted(uppers, conf, side='left') with uppers[i]=fp32(i*fp32(1/15))
// ---------------------------------------------------------------------------
__device__ __forceinline__ void row_stats(const float* __restrict__ p, int label,
                                          int& bin, float& conf, float& corr,
                                          bool& valid)
{
    const float2* q = (const float2*)p;            // 8-byte aligned (offset 40B)
    float2 a = q[0], b = q[1], c = q[2], d = q[3], e = q[4];
    float v[ROW_C] = {a.x, a.y, b.x, b.y, c.x, c.y, d.x, d.y, e.x, e.y};

    float best = v[0];
    int   bi   = 0;
#pragma unroll
    for (int j = 1; j < ROW_C; ++j)
        if (v[j] > best) { best = v[j]; bi = j; }    // strict > keeps first max

    conf  = best;
    valid = best > 0.0f;                             // conf==0 -> dropped bin

    const float step = 1.0f / 15.0f;                 // fp32(1/15), as linspace
    int bn = 0;
#pragma unroll
    for (int i = 1; i <= N_BINS; ++i)
        bn += (step * (float)i < best) ? 1 : 0;      // # uppers strictly < conf
    bin = (bn > N_BINS - 1) ? (N_BINS - 1) : bn;

    corr = (bi == label) ? 1.0f : 0.0f;
}

// ---------------------------------------------------------------------------
// Issue one TDM tile load: 1-D tile of CHUNK_ELTS f32, global -> LDS.
// D# bit layout per CDNA5 ISA 8.3/8.4. Not in a cluster: workgroup_mask=0.
// ---------------------------------------------------------------------------
__device__ __forceinline__ void tdm_issue(uint64_t gaddr, uint32_t lds_off,
                                          const u32x8& g1)
{
    u32x4 g0;
    g0.x = 1u;                                                   // count=1, valid
    g0.y = (uint32_t)__builtin_amdgcn_readfirstlane((int)lds_off);       // lds_addr
    g0.z = (uint32_t)__builtin_amdgcn_readfirstlane((int)(uint32_t)gaddr);
    g0.w = (uint32_t)__builtin_amdgcn_readfirstlane(
               (int)((((uint32_t)(gaddr >> 32)) & 0x01FFFFFFu)   // addr[56:32]
                     | 0x80000000u));                            // type=2 (image)
    asm volatile("tensor_load_to_lds %0, %1" : : "s"(g0), "s"(g1) : "memory");
}

// ---------------------------------------------------------------------------
// Main kernel: each wave streams CPW chunks of 64 rows via TDM double buffer.
// ---------------------------------------------------------------------------
__global__ __launch_bounds__(256) void ece_main(const float* __restrict__ preds,
                                                const int* __restrict__ labels,
                                                float* __restrict__ ws)
{
    __shared__ __align__(16) float s_tile[WAVES_PER_BLOCK][2][CHUNK_ELTS]; // 40KB
    __shared__ float s_acc[48];   // [0..14]=count  [16..30]=Σconf  [32..46]=Σcorr

    const int lane = threadIdx.x & 31;
    const int wv   = threadIdx.x >> 5;

    if (threadIdx.x < 48) s_acc[threadIdx.x] = 0.0f;
    __syncthreads();

    // Constant D# group1: data_size=4B; tensor/tile dim0 = CHUNK_ELTS; dim1 = 1.
    const uint32_t td0 = (uint32_t)CHUNK_ELTS;
    u32x8 g1;
    g1[0] = 2u << 16;                          // workgroup_mask=0, data_size=2(4B)
    g1[1] = (td0 & 0xFFFFu) << 16;             // tensor_dim0[15:0] @ bits 63:48
    g1[2] = (td0 >> 16) | (1u << 16);          // tensor_dim0[31:16] | tensor_dim1=1
    g1[3] = (td0 & 0xFFFFu) << 16;             // tile_dim0 @ bits 127:112
    g1[4] = 1u;                                // tile_dim1=1, tile_dim2=0
    g1[5] = td0;                               // tensor_dim0_stride[31:0]
    g1[6] = (td0 & 0xFFFFu) << 16;             // stride0[47:32]=0 | stride1[15:0]
    g1[7] = 0u;                                // stride1[47:16]

    const long long wave_global = (long long)blockIdx.x * WAVES_PER_BLOCK + wv;
    const long long chunk0      = wave_global * CPW;

    const uint32_t lds_off[2] = {
        (uint32_t)(uintptr_t)(void*)&s_tile[wv][0][0],
        (uint32_t)(uintptr_t)(void*)&s_tile[wv][1][0]
    };
    const uint64_t pbase = (uint64_t)(uintptr_t)preds;

    // Prologue: chunk 0 -> buffer 0.  (TDM ignores EXEC; every wave issues its
    // own tile in straight-line code — no divergent control flow around issues.)
    tdm_issue(pbase + (uint64_t)chunk0 * (CHUNK_ELTS * 4ull), lds_off[0], g1);

#pragma unroll
    for (int i = 0; i < CPW; ++i) {
        const int cur = i & 1;
        if (i + 1 < CPW) {                       // compile-time condition
            tdm_issue(pbase + (uint64_t)(chunk0 + i + 1) * (CHUNK_ELTS * 4ull),
                      lds_off[(i + 1) & 1], g1);
            __builtin_amdgcn_s_wait_tensorcnt(1);  // oldest (chunk i) complete
        } else {
            __builtin_amdgcn_s_wait_tensorcnt(0);  // drain for last chunk
        }

        const float*    tile     = &s_tile[wv][cur][0];
        const long long row_base = (chunk0 + i) * CHUNK_ROWS;
#pragma unroll
        for (int r = 0; r < 2; ++r) {            // 64 rows, 32 lanes, 2 passes
            const int rc = lane + r * 32;        // lane-stride 40B: conflict-free
            int bin; float conf, corr; bool valid;
            row_stats(tile + rc * ROW_C, labels[row_base + rc],
                      bin, conf, corr, valid);
            if (valid) {
                atomicAdd(&s_acc[bin],      1.0f);   // ds_add_f32
                atomicAdd(&s_acc[16 + bin], conf);
                atomicAdd(&s_acc[32 + bin], corr);
            }
        }
    }

    __syncthreads();
    if (threadIdx.x < 48) {
        float v = s_acc[threadIdx.x];
        if (v != 0.0f) unsafeAtomicAdd(&ws[threadIdx.x], v); // global_atomic_add_f32
    }
}

// ---------------------------------------------------------------------------
// Tail kernel: rows not covered by the TDM grid (none for N=4194304).
// ---------------------------------------------------------------------------
__global__ void ece_tail(const float* __restrict__ preds,
                         const int* __restrict__ labels,
                         float* __restrict__ ws,
                         long long row_start, long long n)
{
    long long row = row_start + (long long)blockIdx.x * blockDim.x + threadIdx.x;
    if (row >= n) return;
    int bin; float conf, corr; bool valid;
    row_stats(preds + row * ROW_C, labels[row], bin, conf, corr, valid);
    if (valid) {
        unsafeAtomicAdd(&ws[bin],      1.0f);
        unsafeAtomicAdd(&ws[16 + bin], conf);
        unsafeAtomicAdd(&ws[32 + bin], corr);
    }
}

__global__ void ece_zero(float* __restrict__ ws)
{
    if (threadIdx.x < 48) ws[threadIdx.x] = 0.0f;
}

__global__ void ece_finalize(const float* __restrict__ ws, float* __restrict__ out,
                             long long n)
{
    if (threadIdx.x == 0 && blockIdx.x == 0) {
        const float fn  = (float)n;
        float ece = 0.0f;
        for (int i = 0; i < N_BINS; ++i) {
            float c = ws[i];
            if (c > 0.0f) {
                float avg_conf = ws[16 + i] / c;
                float acc      = ws[32 + i] / c;
                ece += fabsf(avg_conf - acc) * (c / fn);
            }
        }
        out[0] = ece;
    }
}

// ---------------------------------------------------------------------------
extern "C" void kernel_launch(void* const* d_in, const int* in_sizes, int n_in,
                              void* d_out, int out_size, void* d_ws, size_t ws_size,
                              hipStream_t stream)
{
    const float* preds  = (const float*)d_in[0];
    const int*   labels = (const int*)d_in[1];
    float*       ws     = (float*)d_ws;
    float*       out    = (float*)d_out;
    const long long n   = (long long)in_sizes[1];   // labels count = #rows

    ece_zero<<<1, 64, 0, stream>>>(ws);

    const long long blocks  = n / ROWS_PER_BLOCK;   // 2048 for N=4194304
    const long long covered = blocks * ROWS_PER_BLOCK;
    if (blocks > 0)
        ece_main<<<dim3((unsigned)blocks), 256, 0, stream>>>(preds, labels, ws);

    const long long tail = n - covered;
    if (tail > 0) {
        const int tb = (int)((tail + 255) / 256);
        ece_tail<<<tb, 256, 0, stream>>>(preds, labels, ws, covered, n);
    }

    ece_finalize<<<1, 32, 0, stream>>>(ws, out, n);
}